// CBOWModel_18356690223611
// MI455X (gfx1250) — compile-verified
//
#include <hip/hip_runtime.h>

// CBOW negative-sampling loss, MI455X (gfx1250, wave32).
//
// Inputs (setup_inputs order):
//   d_in[0] pos_u    int32 [B, 10]
//   d_in[1] pos_w    int32 [B]
//   d_in[2] neg_w    int32 [B, 5]
//   d_in[3] u_weight f32   [VOCAB, 128]
//   d_in[4] w_weight f32   [VOCAB, 128]
// Output: d_out[0] = -(sum logsig(pos) + sum logsig(-neg))  (f32 scalar)
//
// Design: bandwidth-bound gather (134 MB; tables fit in 192MB L2). One wave
// per 16-row tile; coalesced float4 row gathers -> LDS (stride 132 to kill
// bank conflicts); per-row dots = diag of f32 WMMA 16x16x4 chains.

#define DIM    128
#define CTX    10
#define NEG    5
#define TILE   16
#define STRIDE 132   // 128 + 4 pad: row*132 % 64 banks = row*4 -> conflict-free column reads

typedef __attribute__((ext_vector_type(2))) float v2f;
typedef __attribute__((ext_vector_type(8))) float v8f;

__device__ __forceinline__ float logsig(float x) {
    // log(sigmoid(x)) = min(x,0) - log1p(exp(-|x|)), numerically stable
    return fminf(x, 0.0f) - log1pf(expf(-fabsf(x)));
}

__global__ __launch_bounds__(32)
void cbow_tile_kernel(const int* __restrict__ pos_u,
                      const int* __restrict__ pos_w,
                      const int* __restrict__ neg_w,
                      const float* __restrict__ u_weight,
                      const float* __restrict__ w_weight,
                      float* __restrict__ partials) {
    __shared__ float Ul[TILE * STRIDE];
    __shared__ float Wpl[TILE * STRIDE];
    __shared__ float Wnl[TILE * STRIDE];
    __shared__ float Dp[16 * 16];
    __shared__ float Dn[16 * 16];

    const int lane  = threadIdx.x;        // 0..31
    const int lane4 = lane << 2;          // float4 chunk of the 128-dim row
    const int base  = blockIdx.x * TILE;  // first batch row of this tile

    // ---------------- Phase 1: coalesced gathers -> LDS ----------------
    // One row at a time; all 32 lanes cooperate (512B per global_load_b128 op).
    for (int r = 0; r < TILE; ++r) {
        const int b = base + r;

        float u0 = 0.f, u1 = 0.f, u2 = 0.f, u3 = 0.f;
        #pragma unroll
        for (int j = 0; j < CTX; ++j) {
            const int idx = pos_u[b * CTX + j];                 // uniform -> SMEM load
            const float4 v = *reinterpret_cast<const float4*>(
                u_weight + (size_t)idx * DIM + lane4);
            u0 += v.x; u1 += v.y; u2 += v.z; u3 += v.w;
        }

        const int pidx = pos_w[b];
        const float4 wp = *reinterpret_cast<const float4*>(
            w_weight + (size_t)pidx * DIM + lane4);

        float n0 = 0.f, n1 = 0.f, n2 = 0.f, n3 = 0.f;
        #pragma unroll
        for (int j = 0; j < NEG; ++j) {
            const int idx = neg_w[b * NEG + j];
            const float4 v = *reinterpret_cast<const float4*>(
                w_weight + (size_t)idx * DIM + lane4);
            n0 += v.x; n1 += v.y; n2 += v.z; n3 += v.w;
        }

        float4 uu; uu.x = u0; uu.y = u1; uu.z = u2; uu.w = u3;
        float4 nn; nn.x = n0; nn.y = n1; nn.z = n2; nn.w = n3;
        *reinterpret_cast<float4*>(&Ul [r * STRIDE + lane4]) = uu;  // ds_store_b128
        *reinterpret_cast<float4*>(&Wpl[r * STRIDE + lane4]) = wp;
        *reinterpret_cast<float4*>(&Wnl[r * STRIDE + lane4]) = nn;
    }
    __syncthreads();

    // ---------------- Phase 2: dual WMMA f32 16x16x4 chains ----------------
    // A[m][k] = U[m][k0+k]; B[k][n] = W[n][k0+k]  =>  D[m][n] = pairwise dots.
    // ISA 16x4 f32 lane layout: lanes 0-15 hold K0(V0)/K1(V1), lanes 16-31 K2/K3.
    const int row16 = lane & 15;
    const int koff  = (lane >> 4) << 1;   // 0 for lanes 0-15, 2 for lanes 16-31

    v8f accp = {};
    v8f accn = {};
    #pragma unroll
    for (int k0 = 0; k0 < DIM; k0 += 4) {
        const v2f a  = *reinterpret_cast<const v2f*>(&Ul [row16 * STRIDE + k0 + koff]);
        const v2f bp = *reinterpret_cast<const v2f*>(&Wpl[row16 * STRIDE + k0 + koff]);
        const v2f bn = *reinterpret_cast<const v2f*>(&Wnl[row16 * STRIDE + k0 + koff]);
        accp = __builtin_amdgcn_wmma_f32_16x16x4_f32(
            false, a, false, bp, (short)0, accp, false, false);
        accn = __builtin_amdgcn_wmma_f32_16x16x4_f32(
            false, a, false, bn, (short)0, accn, false, false);
    }

    // ---------------- Diagonal extraction via LDS spill ----------------
    // C/D layout: VGPR i, lanes 0-15 -> (M=i, N=lane); lanes 16-31 -> (M=i+8, N=lane-16)
    const int mbase = (lane >> 4) << 3;
    const int ncol  = lane & 15;
    #pragma unroll
    for (int i = 0; i < 8; ++i) {
        Dp[(mbase + i) * 16 + ncol] = accp[i];
        Dn[(mbase + i) * 16 + ncol] = accn[i];
    }
    __syncthreads();

    float t = 0.0f;
    if (lane < 16) {
        const float sp = Dp[lane * 17];   // diag: M == N == lane
        const float sn = Dn[lane * 17];
        t = logsig(sp) + logsig(-sn);
    }
    // wave32 reduction (lanes 16-31 contribute 0)
    #pragma unroll
    for (int m = 16; m > 0; m >>= 1) t += __shfl_xor(t, m, 32);

    if (lane == 0) partials[blockIdx.x] = t;
}

__global__ __launch_bounds__(256)
void cbow_reduce_kernel(const float* __restrict__ partials, float* __restrict__ out,
                        int nblocks) {
    __shared__ float red[256];
    float s = 0.0f;
    for (int i = threadIdx.x; i < nblocks; i += 256) s += partials[i];  // fixed order
    red[threadIdx.x] = s;
    __syncthreads();
    if (threadIdx.x == 0) {
        float tot = 0.0f;
        for (int i = 0; i < 256; ++i) tot += red[i];                    // fixed order
        out[0] = -tot;   // reference returns -loss
    }
}

extern "C" void kernel_launch(void* const* d_in, const int* in_sizes, int n_in,
                              void* d_out, int out_size, void* d_ws, size_t ws_size,
                              hipStream_t stream) {
    const int*   pos_u    = (const int*)  d_in[0];
    const int*   pos_w    = (const int*)  d_in[1];
    const int*   neg_w    = (const int*)  d_in[2];
    const float* u_weight = (const float*)d_in[3];
    const float* w_weight = (const float*)d_in[4];

    const int B       = in_sizes[1];      // pos_w is [B]
    const int nblocks = B / TILE;         // 16384/16 = 1024

    float* partials = (float*)d_ws;       // nblocks * 4 bytes of scratch
    float* out      = (float*)d_out;

    cbow_tile_kernel<<<nblocks, 32, 0, stream>>>(
        pos_u, pos_w, neg_w, u_weight, w_weight, partials);
    cbow_reduce_kernel<<<1, 256, 0, stream>>>(partials, out, nblocks);
}